// HypergraphLayer_68143951118560
// MI455X (gfx1250) — compile-verified
//
#include <hip/hip_runtime.h>
#include <hip/hip_bf16.h>

typedef __attribute__((ext_vector_type(2))) float v2f;
typedef __attribute__((ext_vector_type(8))) float v8f;

#define F 128
#define BN_EPS 1e-5f

// ---------------------------------------------------------------------------
// Kernel 1: degree histograms. D[i] = sum of w[col] over incidences at node i,
//           B[e] = #incidences at hyperedge e.
// ---------------------------------------------------------------------------
__global__ __launch_bounds__(256) void degrees_kernel(
    const int* __restrict__ row, const int* __restrict__ col,
    const float* __restrict__ w, float* __restrict__ D, float* __restrict__ B,
    int E) {
  int e = blockIdx.x * blockDim.x + threadIdx.x;
  if (e < E) {
    int r = row[e];
    int c = col[e];
    atomicAdd(&D[r], w[c]);
    atomicAdd(&B[c], 1.0f);
  }
}

// ---------------------------------------------------------------------------
// Kernel 2: H = X @ W via V_WMMA_F32_16X16X4_F32. One wave per 16x16 output
// tile, K=128 in 32 steps of 4. EXEC stays all-1s throughout the WMMA loop
// (row loads clamped). Store epilogue: single uniform branch for full tiles
// (the common case -> 8 straight global_store_b32), guarded loop only on the
// tail tile.
// ---------------------------------------------------------------------------
__global__ __launch_bounds__(32) void gemm_wmma_f32(
    const float* __restrict__ X, const float* __restrict__ W,
    float* __restrict__ H, int Nrows) {
  const int tileM = blockIdx.x * 16;
  const int tileN = blockIdx.y * 16;
  const int lane  = threadIdx.x;      // 0..31
  const int half  = lane >> 4;        // 0 or 1
  const int l     = lane & 15;

  int rowIdx = tileM + l;
  if (rowIdx > Nrows - 1) rowIdx = Nrows - 1;  // clamp, keep EXEC full

  v8f acc = {};
  const float* xp0 = X + (long)rowIdx * F;
#pragma unroll 4
  for (int k0 = 0; k0 < F; k0 += 4) {
    int ka = k0 + 2 * half;  // this half-wave's K pair: {ka, ka+1}
    v2f a, b;
    a.x = xp0[ka];
    a.y = xp0[ka + 1];
    const float* wp = W + ka * F + tileN + l;
    b.x = wp[0];
    b.y = wp[F];
    // 8 args: (neg_a, A, neg_b, B, c_mod, C, reuse_a, reuse_b)
    acc = __builtin_amdgcn_wmma_f32_16x16x4_f32(
        false, a, false, b, (short)0, acc, false, false);
  }

  float* out0 = H + (long)(tileM + 8 * half) * F + tileN + l;
  if (tileM + 16 <= Nrows) {
    // Full tile (always the case when Nrows % 16 == 0): straight stores.
#pragma unroll
    for (int r = 0; r < 8; ++r) {
      out0[(long)r * F] = acc[r];
    }
  } else {
    // Tail tile: per-row guard.
#pragma unroll
    for (int r = 0; r < 8; ++r) {
      int m = tileM + r + 8 * half;
      if (m < Nrows) H[(long)m * F + tileN + l] = acc[r];
    }
  }
}

// ---------------------------------------------------------------------------
// Kernel 3: node -> hyperedge scatter. One wave per incidence: coalesced
// float4 gather of h[row], 4 f32 atomics per lane into e_feat[col].
// ---------------------------------------------------------------------------
__global__ __launch_bounds__(256) void scatter_n2e(
    const int* __restrict__ row, const int* __restrict__ col,
    const float* __restrict__ H, float* __restrict__ Efeat, int E) {
  int wave = (blockIdx.x * blockDim.x + threadIdx.x) >> 5;
  int lane = threadIdx.x & 31;
  if (wave >= E) return;
  int r = row[wave];
  int c = col[wave];
  const float4 hv = *(const float4*)(H + (long)r * F + lane * 4);
  float* dst = Efeat + (long)c * F + lane * 4;
  atomicAdd(dst + 0, hv.x);
  atomicAdd(dst + 1, hv.y);
  atomicAdd(dst + 2, hv.z);
  atomicAdd(dst + 3, hv.w);
}

// ---------------------------------------------------------------------------
// Kernel 4: hyperedge -> node scatter, folding B_inv[col] into the gather.
// Accumulates into d_out (zeroed beforehand).
// ---------------------------------------------------------------------------
__global__ __launch_bounds__(256) void scatter_e2n(
    const int* __restrict__ row, const int* __restrict__ col,
    const float* __restrict__ Efeat, const float* __restrict__ B,
    float* __restrict__ Acc, int E) {
  int wave = (blockIdx.x * blockDim.x + threadIdx.x) >> 5;
  int lane = threadIdx.x & 31;
  if (wave >= E) return;
  int r = row[wave];
  int c = col[wave];
  float bd   = B[c];
  float binv = bd > 0.0f ? 1.0f / bd : 0.0f;
  const float4 ev = *(const float4*)(Efeat + (long)c * F + lane * 4);
  float* dst = Acc + (long)r * F + lane * 4;
  atomicAdd(dst + 0, ev.x * binv);
  atomicAdd(dst + 1, ev.y * binv);
  atomicAdd(dst + 2, ev.z * binv);
  atomicAdd(dst + 3, ev.w * binv);
}

// ---------------------------------------------------------------------------
// Kernel 5: out = D_inv * acc + bias (in place on d_out), plus per-feature
// sum / sumsq partial reduction (block-local accumulate, 2 atomics/feature).
// blockDim = 128 (one thread per feature), FIN_ROWS rows per block.
// ---------------------------------------------------------------------------
#define FIN_ROWS 64
__global__ __launch_bounds__(F) void finalize_stats(
    const float* __restrict__ D, const float* __restrict__ bias,
    float* __restrict__ Out, float* __restrict__ Sum,
    float* __restrict__ Sumsq, int Nn) {
  int f  = threadIdx.x;
  int r0 = blockIdx.x * FIN_ROWS;
  float s = 0.0f, sq = 0.0f;
  float bf = bias[f];
  for (int i = 0; i < FIN_ROWS; ++i) {
    int r = r0 + i;
    if (r >= Nn) break;
    float d    = D[r];
    float dinv = d > 0.0f ? 1.0f / d : 0.0f;
    long idx   = (long)r * F + f;
    float v    = Out[idx] * dinv + bf;
    Out[idx]   = v;
    s  += v;
    sq += v * v;
  }
  atomicAdd(&Sum[f], s);
  atomicAdd(&Sumsq[f], sq);
}

// ---------------------------------------------------------------------------
// Kernel 6: BatchNorm (batch stats, biased var) + ReLU, in place on d_out.
// ---------------------------------------------------------------------------
__global__ __launch_bounds__(256) void bn_relu(
    float* __restrict__ Out, const float* __restrict__ Sum,
    const float* __restrict__ Sumsq, const float* __restrict__ gamma,
    const float* __restrict__ beta, int Nn) {
  long idx = (long)blockIdx.x * blockDim.x + threadIdx.x;
  long total = (long)Nn * F;
  if (idx >= total) return;
  int f = (int)(idx & (F - 1));
  float invN  = 1.0f / (float)Nn;
  float mean  = Sum[f] * invN;
  float var   = Sumsq[f] * invN - mean * mean;
  float scale = gamma[f] * rsqrtf(var + BN_EPS);
  float v     = scale * (Out[idx] - mean) + beta[f];
  Out[idx]    = v > 0.0f ? v : 0.0f;
}

// ---------------------------------------------------------------------------
// Host launcher
// ---------------------------------------------------------------------------
extern "C" void kernel_launch(void* const* d_in, const int* in_sizes, int n_in,
                              void* d_out, int out_size, void* d_ws,
                              size_t ws_size, hipStream_t stream) {
  const float* x     = (const float*)d_in[0];
  const int*   hidx  = (const int*)d_in[1];
  const float* hw    = (const float*)d_in[2];
  const float* W     = (const float*)d_in[3];
  const float* bias  = (const float*)d_in[4];
  const float* gamma = (const float*)d_in[5];
  const float* beta  = (const float*)d_in[6];

  const int N = in_sizes[0] / F;       // nodes
  const int E = in_sizes[1] / 2;       // incidences
  const int M = in_sizes[2];           // hyperedges

  const int* rowIdx = hidx;            // hyperedge_index[0]
  const int* colIdx = hidx + E;        // hyperedge_index[1]

  // Workspace layout (floats)
  float* ws    = (float*)d_ws;
  float* D     = ws;                               // N
  float* B     = D + N;                            // M
  float* H     = B + M;                            // N*F
  float* Efeat = H + (size_t)N * F;                // M*F
  float* Sum   = Efeat + (size_t)M * F;            // F
  float* Sumsq = Sum + F;                          // F
  float* Out   = (float*)d_out;                    // N*F (accumulator + result)

  // Zero all accumulators (capture-safe async memsets)
  hipMemsetAsync(D, 0, sizeof(float) * (size_t)N, stream);
  hipMemsetAsync(B, 0, sizeof(float) * (size_t)M, stream);
  hipMemsetAsync(Efeat, 0, sizeof(float) * (size_t)M * F, stream);
  hipMemsetAsync(Sum, 0, sizeof(float) * 2 * F, stream);
  hipMemsetAsync(Out, 0, sizeof(float) * (size_t)N * F, stream);

  // 1) degrees
  degrees_kernel<<<(E + 255) / 256, 256, 0, stream>>>(rowIdx, colIdx, hw, D, B, E);

  // 2) H = X @ W  (WMMA f32 16x16x4)
  dim3 ggrid((N + 15) / 16, F / 16);
  gemm_wmma_f32<<<ggrid, 32, 0, stream>>>(x, W, H, N);

  // 3) node -> hyperedge (one wave per incidence)
  int waves_per_block = 256 / 32;
  scatter_n2e<<<(E + waves_per_block - 1) / waves_per_block, 256, 0, stream>>>(
      rowIdx, colIdx, H, Efeat, E);

  // 4) hyperedge -> node with B_inv folded in
  scatter_e2n<<<(E + waves_per_block - 1) / waves_per_block, 256, 0, stream>>>(
      rowIdx, colIdx, Efeat, B, Out, E);

  // 5) D_inv * acc + bias, and per-feature sum/sumsq
  finalize_stats<<<(N + FIN_ROWS - 1) / FIN_ROWS, F, 0, stream>>>(
      D, bias, Out, Sum, Sumsq, N);

  // 6) BatchNorm + ReLU
  long total = (long)N * F;
  bn_relu<<<(int)((total + 255) / 256), 256, 0, stream>>>(
      Out, Sum, Sumsq, gamma, beta, N);
}